// SSIM_Derain_GPU_34265249087971
// MI455X (gfx1250) — compile-verified
//
#include <hip/hip_runtime.h>
#include <math.h>

// ---------------------------------------------------------------------------
// Fused SSIM (11x11 Gaussian, sigma=1.5) for 1x3x4096x4096 fp32 image pairs.
// Single pass over HBM; both separable blur passes run on the CDNA5 matrix
// pipe as banded-matrix products using V_WMMA_F32_16X16X4_F32 (fp32, matching
// reference precision). Wave-level work ids are forced scalar so EXEC remains
// all-ones through every WMMA (ISA requirement) and control flow is uniform.
// Final mean: partials spread over 1024 slots (avoids same-address atomic
// serialization), then a tiny reduction kernel produces the scalar.
// ---------------------------------------------------------------------------

typedef __attribute__((ext_vector_type(2))) float v2f;
typedef __attribute__((ext_vector_type(8))) float v8f;

#define TILE   32      // output tile (rows == cols) per block
#define RAW_R  48      // raw LDS rows: tile rows [-5 .. 42]
#define RAW_C  49      // 48 cols [-5 .. 42] + 1 pad
#define INT_C  33      // 32 intermediate cols + 1 pad
#define NTHREADS 128   // 4 wave32s
#define NSLOTS 1024    // partial-sum accumulator slots in d_ws

// Normalized 1D Gaussian, ws=11, sigma=1.5 (outer product == reference window)
static __device__ __constant__ float CW[11] = {
    0.00102838f, 0.00759876f, 0.03600077f, 0.10936069f, 0.21300553f,
    0.26601172f,
    0.21300553f, 0.10936069f, 0.03600077f, 0.00759876f, 0.00102838f
};

__device__ __forceinline__ float quant255(float t) {
    // reference: floor(clip((t+1)/2*255, 0, 255))
    float u = (t + 1.0f) * 0.5f * 255.0f;
    return floorf(fminf(fmaxf(u, 0.0f), 255.0f));
}

__global__ __launch_bounds__(NTHREADS)
void ssim_fused_kernel(const float* __restrict__ img1,
                       const float* __restrict__ img2,
                       float* __restrict__ partial,   // NSLOTS accumulators (or 1)
                       int nslots,
                       int H, int W)
{
    __shared__ float lg1[RAW_R * RAW_C];     // g1 tile (zero padded)
    __shared__ float lg2[RAW_R * RAW_C];     // g2 tile (zero padded)
    __shared__ float li[5][RAW_R * INT_C];   // horizontally blurred channels
    __shared__ float wpad[64];               // zero-padded Gaussian band

    const int HW  = H * W;
    const int tid = threadIdx.x;
    const int ty0 = blockIdx.y * TILE;
    const int tx0 = blockIdx.x * TILE;

    // ---------------- Stage 0: padded weight table ----------------
    if (tid < 64) {
        const int wi = tid - 16;
        wpad[tid] = (wi >= 0 && wi < 11) ? CW[wi] : 0.0f;
    }

    // ---------------- Stage 1: load + RGB->Y, zero pad ----------------
    for (int p = tid; p < RAW_R * RAW_C; p += NTHREADS) {
        const int i = p / RAW_C;        // tile row + 5
        const int j = p - i * RAW_C;    // tile col + 5
        float a = 0.0f, b = 0.0f;
        const int gy = ty0 + i - 5;
        const int gx = tx0 + j - 5;
        if (i < TILE + 10 && j < TILE + 10 &&
            (unsigned)gy < (unsigned)H && (unsigned)gx < (unsigned)W) {
            const int base = gy * W + gx;
            {
                float r  = quant255(img1[base]);
                float g  = quant255(img1[HW + base]);
                float bl = quant255(img1[2 * HW + base]);
                float y  = r * 0.299f + g * 0.587f + bl * 0.114f;
                a = (y * (219.0f / 255.0f) + 16.0f) * (1.0f / 255.0f);
            }
            {
                float r  = quant255(img2[base]);
                float g  = quant255(img2[HW + base]);
                float bl = quant255(img2[2 * HW + base]);
                float y  = r * 0.299f + g * 0.587f + bl * 0.114f;
                b = (y * (219.0f / 255.0f) + 16.0f) * (1.0f / 255.0f);
            }
        }
        lg1[p] = a;
        lg2[p] = b;
    }
    __syncthreads();

    // Scalarize the wave id: uniform control flow, EXEC stays full for WMMA.
    const int wave = __builtin_amdgcn_readfirstlane(tid >> 5);
    const int lane = tid & 31;
    const int mn   = lane & 15;   // A row (M) == B/D col (N) for this lane
    const int half = lane >> 4;   // K-half per ISA 32-bit 16x4 layout

    // Per-lane Gaussian band, shared by horizontal B and vertical A:
    // wband[2k+v] = w[(4k + 2*half + v) - mn] (zero outside the band).
    float wband[14];
    #pragma unroll
    for (int k = 0; k < 7; ++k) {
        #pragma unroll
        for (int v = 0; v < 2; ++v) {
            const int kap = 4 * k + 2 * half + v;
            wband[2 * k + v] = wpad[kap - mn + 16];
        }
    }

    // ---------------- Stage 2: horizontal blur via WMMA ----------------
    // Task = (rowgroup, colblock); each task runs all 5 channels so the 28
    // LDS input loads are amortized over 5 WMMA chains.
    for (int t = wave; t < 6; t += 4) {
        const int rowbase = (t >> 1) * 16;   // raw row base: 0,16,32
        const int cb      = (t & 1) * 16;    // output col block: 0,16

        // Per-lane input values for the 7 K-chunks (A-matrix side).
        const int abase = (rowbase + mn) * RAW_C + cb;
        float x1[14], x2[14];
        #pragma unroll
        for (int k = 0; k < 7; ++k) {
            #pragma unroll
            for (int v = 0; v < 2; ++v) {
                const int kap = 4 * k + 2 * half + v;
                x1[2 * k + v] = lg1[abase + kap];
                x2[2 * k + v] = lg2[abase + kap];
            }
        }

        v8f acc[5];
        #pragma unroll
        for (int c = 0; c < 5; ++c) acc[c] = (v8f){0.f,0.f,0.f,0.f,0.f,0.f,0.f,0.f};

        #pragma unroll
        for (int k = 0; k < 7; ++k) {
            const v2f B = { wband[2 * k], wband[2 * k + 1] };
            const float a10 = x1[2 * k], a11 = x1[2 * k + 1];
            const float a20 = x2[2 * k], a21 = x2[2 * k + 1];
            v2f A;
            A = (v2f){ a10, a11 };
            acc[0] = __builtin_amdgcn_wmma_f32_16x16x4_f32(false, A, false, B, (short)0, acc[0], false, false);
            A = (v2f){ a20, a21 };
            acc[1] = __builtin_amdgcn_wmma_f32_16x16x4_f32(false, A, false, B, (short)0, acc[1], false, false);
            A = (v2f){ a10 * a10, a11 * a11 };
            acc[2] = __builtin_amdgcn_wmma_f32_16x16x4_f32(false, A, false, B, (short)0, acc[2], false, false);
            A = (v2f){ a20 * a20, a21 * a21 };
            acc[3] = __builtin_amdgcn_wmma_f32_16x16x4_f32(false, A, false, B, (short)0, acc[3], false, false);
            A = (v2f){ a10 * a20, a11 * a21 };
            acc[4] = __builtin_amdgcn_wmma_f32_16x16x4_f32(false, A, false, B, (short)0, acc[4], false, false);
        }

        const int sbase = rowbase * INT_C + cb + mn;
        #pragma unroll
        for (int c = 0; c < 5; ++c) {
            #pragma unroll
            for (int v = 0; v < 8; ++v) {
                const int m = v + 8 * half;          // D row per ISA C/D layout
                li[c][sbase + m * INT_C] = acc[c][v];
            }
        }
    }
    __syncthreads();

    // ---------------- Stage 3: vertical blur via WMMA + SSIM ----------------
    // Out(16x16) = sum_k Wband(16x4) x Interm(4x16); one 16x16 tile per wave.
    {
        const int rb = (wave >> 1) * 16;   // output row block (scalar)
        const int cb = (wave & 1) * 16;    // output col block (scalar)

        v8f acc[5];
        #pragma unroll
        for (int c = 0; c < 5; ++c) acc[c] = (v8f){0.f,0.f,0.f,0.f,0.f,0.f,0.f,0.f};

        const int bbase = rb * INT_C + cb + mn;
        #pragma unroll
        for (int k = 0; k < 7; ++k) {
            const v2f A = { wband[2 * k], wband[2 * k + 1] };  // w[kap - m], m = mn
            const int k0 = 4 * k + 2 * half;
            #pragma unroll
            for (int c = 0; c < 5; ++c) {
                const v2f B = { li[c][bbase + k0 * INT_C],
                                li[c][bbase + (k0 + 1) * INT_C] };
                acc[c] = __builtin_amdgcn_wmma_f32_16x16x4_f32(
                    false, A, false, B, (short)0, acc[c], false, false);
            }
        }

        // SSIM map + mean contribution
        const float C1 = 1e-4f, C2 = 9e-4f;
        float s = 0.0f;
        #pragma unroll
        for (int v = 0; v < 8; ++v) {
            const int m  = v + 8 * half;
            const int oy = ty0 + rb + m;
            const int ox = tx0 + cb + mn;
            if (oy < H && ox < W) {
                const float mu1 = acc[0][v], mu2 = acc[1][v];
                const float m11 = mu1 * mu1, m22 = mu2 * mu2, m12 = mu1 * mu2;
                const float s11 = acc[2][v] - m11;
                const float s22 = acc[3][v] - m22;
                const float s12 = acc[4][v] - m12;
                s += __fdividef((2.0f * m12 + C1) * (2.0f * s12 + C2),
                                (m11 + m22 + C1) * (s11 + s22 + C2));
            }
        }
        s *= (1.0f / (float)HW);

        // wave32 reduction, one atomic per wave, spread across slots
        #pragma unroll
        for (int off = 16; off >= 1; off >>= 1)
            s += __shfl_xor(s, off);
        if (lane == 0) {
            const int slot = (blockIdx.y * gridDim.x + blockIdx.x) & (nslots - 1);
            atomicAdd(&partial[slot], s);
        }
    }
}

__global__ __launch_bounds__(256)
void ssim_reduce_kernel(const float* __restrict__ partial, float* __restrict__ out,
                        int n)
{
    __shared__ float red[256];
    float s = 0.0f;
    for (int i = threadIdx.x; i < n; i += 256) s += partial[i];
    red[threadIdx.x] = s;
    __syncthreads();
    #pragma unroll
    for (int off = 128; off >= 1; off >>= 1) {
        if ((int)threadIdx.x < off) red[threadIdx.x] += red[threadIdx.x + off];
        __syncthreads();
    }
    if (threadIdx.x == 0) out[0] = red[0];
}

extern "C" void kernel_launch(void* const* d_in, const int* in_sizes, int n_in,
                              void* d_out, int out_size, void* d_ws, size_t ws_size,
                              hipStream_t stream) {
    const float* img1 = (const float*)d_in[0];
    const float* img2 = (const float*)d_in[1];
    float* out = (float*)d_out;

    const int HW3 = in_sizes[0];
    const int HW  = HW3 / 3;
    int W = (int)(sqrt((double)HW) + 0.5);
    if (W <= 0) W = 1;
    const int H = HW / W;

    dim3 grid((W + TILE - 1) / TILE, (H + TILE - 1) / TILE);

    if (ws_size >= NSLOTS * sizeof(float)) {
        float* partial = (float*)d_ws;
        hipMemsetAsync(partial, 0, NSLOTS * sizeof(float), stream);
        ssim_fused_kernel<<<grid, NTHREADS, 0, stream>>>(img1, img2, partial,
                                                         NSLOTS, H, W);
        ssim_reduce_kernel<<<1, 256, 0, stream>>>(partial, out, NSLOTS);
    } else {
        // Fallback: accumulate directly into the output scalar.
        hipMemsetAsync(d_out, 0, sizeof(float), stream);
        ssim_fused_kernel<<<grid, NTHREADS, 0, stream>>>(img1, img2, out, 1, H, W);
    }
}